// LinearAttention_64811056497155
// MI455X (gfx1250) — compile-verified
//
#include <hip/hip_runtime.h>
#include <hip/hip_bf16.h>
#include <math.h>

// ---------------------------------------------------------------------------
// LinearAttention (B=16,H=64,W=64,C=192,NH=6) for gfx1250 (MI455X, wave32).
// All matrix work routed through V_WMMA_F32_16X16X4_F32 (exact fp32 path).
//
// Workspace layout (floats):
//   cosT : 64*64*96        =   393216
//   sinT : 64*64*96        =   393216
//   Qe   : 16*4096*192     = 12582912   (elu(q)+1)
//   Ke   : 16*4096*192     = 12582912   (elu(k)+1)
//   ksum : 16*192          =     3072   (sum_n Ke, per (b,c))
//   kv   : 16*6*32*32      =    98304   (per-head kv accumulators)
// total ~104 MB.
// ---------------------------------------------------------------------------

#define Bb   16
#define Hh   64
#define Ww   64
#define Cc   192
#define NHh  6
#define Dd   32
#define Nn   4096

typedef __attribute__((ext_vector_type(2))) float v2f;
typedef __attribute__((ext_vector_type(8))) float v8f;

__device__ __forceinline__ v8f wmma_f32(v2f a, v2f b, v8f c) {
  return __builtin_amdgcn_wmma_f32_16x16x4_f32(false, a, false, b,
                                               (short)0, c, false, false);
}

// ---------------------------------------------------------------------------
// K0: RoPE tables + zero the float accumulators (required every call since
// the graph replays without re-poisoning and we accumulate with atomics).
// ---------------------------------------------------------------------------
__global__ void __launch_bounds__(256) init_kernel(float* __restrict__ cosT,
                                                   float* __restrict__ sinT,
                                                   float* __restrict__ ksum,
                                                   float* __restrict__ kvb) {
  int idx = blockIdx.x * 256 + threadIdx.x;
  if (idx < Hh * Ww * 96) {
    int j = idx % 96;
    int n = idx / 96;
    int hp = n >> 6, wp = n & 63;
    int t = (j < 48) ? j : j - 48;
    float pos = (j < 48) ? (float)hp : (float)wp;
    // theta = 10000^(-t/48)
    float theta = __expf(-(float)t * (logf(10000.0f) / 48.0f));
    float ang = pos * theta;
    float s, c;
    __sincosf(ang, &s, &c);
    cosT[idx] = c;
    sinT[idx] = s;
  }
  if (idx < Bb * Cc) ksum[idx] = 0.0f;
  if (idx < Bb * NHh * Dd * Dd) kvb[idx] = 0.0f;
}

// ---------------------------------------------------------------------------
// K1: qk = x @ qk_w^T + qk_b, fused elu(.)+1, split into Qe / Ke.
// Block tile 64(M) x 64(Ncol); 8 waves = 4(M) x 2(N); wave tile 16x32.
// M = 65536 rows, K = 192, Ncol = 384.
// ---------------------------------------------------------------------------
__global__ void __launch_bounds__(256) qk_gemm_kernel(
    const float* __restrict__ x, const float* __restrict__ qk_w,
    const float* __restrict__ qk_b, float* __restrict__ Qe,
    float* __restrict__ Ke) {
  const int tid = threadIdx.x;
  const int wave = tid >> 5;
  const int lane = tid & 31;
  const int l16 = lane & 15;
  const int lhi = lane >> 4;  // 0/1
  const int wm = wave & 3;    // M wave 0..3
  const int wn = wave >> 2;   // N wave 0..1
  const long mbase = (long)blockIdx.x * 64 + wm * 16;
  const int col0 = blockIdx.y * 64 + wn * 32;

  v8f acc0 = {};
  v8f acc1 = {};
  const float* arow = x + (mbase + l16) * Cc;        // A row, K contiguous
  const float* b0row = qk_w + (long)(col0 + l16) * Cc;
  const float* b1row = qk_w + (long)(col0 + 16 + l16) * Cc;

#pragma unroll 4
  for (int k0 = 0; k0 < Cc; k0 += 4) {
    const int ka = k0 + 2 * lhi;
    v2f a = *(const v2f*)(arow + ka);   // A[m][ka], A[m][ka+1]
    v2f b0 = *(const v2f*)(b0row + ka); // B[ka][n] = qk_w[n][ka]
    v2f b1 = *(const v2f*)(b1row + ka);
    acc0 = wmma_f32(a, b0, acc0);
    acc1 = wmma_f32(a, b1, acc1);
  }

#pragma unroll
  for (int t = 0; t < 2; ++t) {
    v8f acc = t ? acc1 : acc0;
    const int col = col0 + t * 16 + l16;
    const float bias = qk_b[col];
    float* dst = (col < Cc) ? Qe : Ke;
    const int ccol = (col < Cc) ? col : col - Cc;
#pragma unroll
    for (int r = 0; r < 8; ++r) {
      const long row = mbase + r + 8 * lhi;
      float v = acc[r] + bias;
      v = (v > 0.0f) ? (v + 1.0f) : __expf(v);  // elu(v)+1
      dst[row * Cc + ccol] = v;
    }
  }
}

// rotated element c of rope(vec) at position n (table indexed by n*96+j)
__device__ __forceinline__ float rope_rot(const float* __restrict__ base,
                                          const float* __restrict__ cosT,
                                          const float* __restrict__ sinT,
                                          int n, int c) {
  const int j = c >> 1;
  v2f eo = *(const v2f*)(base + (long)n * Cc + (c & ~1));
  const float cs = cosT[n * 96 + j];
  const float sn = sinT[n * 96 + j];
  return (c & 1) ? (eo.x * sn + eo.y * cs) : (eo.x * cs - eo.y * sn);
}

// ---------------------------------------------------------------------------
// K2: per (b,h): kv[d][e] = sum_n rope(Ke)[n][d] * x[n][e] / 4096
//     and ksum[b, h*32+d] = sum_n Ke[n][d].
// grid = (96 bh, 8 n-slices of 512). 8 waves each take 64 n (16 WMMA iters),
// computing the full 2x2 set of 16x16 tiles; LDS ds_add_f32 reduce, then
// global atomicAdd into kv.
// ---------------------------------------------------------------------------
__global__ void __launch_bounds__(256) kv_kernel(
    const float* __restrict__ x, const float* __restrict__ Ke,
    const float* __restrict__ cosT, const float* __restrict__ sinT,
    float* __restrict__ ksum, float* __restrict__ kv) {
  __shared__ float kvLds[Dd * Dd];
  __shared__ float ksumLds[Dd];

  const int tid = threadIdx.x;
  const int bh = blockIdx.x;
  const int slice = blockIdx.y;
  const int b = bh / NHh, h = bh % NHh;
  const int c0 = h * Dd;
  const float* KeB = Ke + (long)b * Nn * Cc;
  const float* xB = x + (long)b * Nn * Cc;

  for (int i = tid; i < Dd * Dd; i += 256) kvLds[i] = 0.0f;
  if (tid < Dd) ksumLds[tid] = 0.0f;
  __syncthreads();

  // ---- ksum partials (unrotated Ke) ----
  {
    float s[Dd];
#pragma unroll
    for (int i = 0; i < Dd; ++i) s[i] = 0.0f;
#pragma unroll
    for (int rep = 0; rep < 2; ++rep) {
      const int n = slice * 512 + rep * 256 + tid;
      const float* p = KeB + (long)n * Cc + c0;
#pragma unroll
      for (int i = 0; i < Dd; ++i) s[i] += p[i];
    }
#pragma unroll
    for (int i = 0; i < Dd; ++i) atomicAdd(&ksumLds[i], s[i]);
  }

  // ---- kv via WMMA: M=d (2 tiles), N=e (2 tiles), K=n ----
  const int wave = tid >> 5, lane = tid & 31, l16 = lane & 15, lhi = lane >> 4;
  const float s2 = 1.0f / (float)Nn;  // scale^2, scale = 1/sqrt(n)
  v8f a00 = {}, a01 = {}, a10 = {}, a11 = {};
  const int nbase = slice * 512 + wave * 64;

  for (int it = 0; it < 16; ++it) {
    const int na = nbase + it * 4 + 2 * lhi;
    v2f A0, A1, B0, B1;
    A0.x = rope_rot(KeB, cosT, sinT, na,     c0 + l16) * s2;
    A0.y = rope_rot(KeB, cosT, sinT, na + 1, c0 + l16) * s2;
    A1.x = rope_rot(KeB, cosT, sinT, na,     c0 + 16 + l16) * s2;
    A1.y = rope_rot(KeB, cosT, sinT, na + 1, c0 + 16 + l16) * s2;
    B0.x = xB[(long)na * Cc + c0 + l16];
    B0.y = xB[(long)(na + 1) * Cc + c0 + l16];
    B1.x = xB[(long)na * Cc + c0 + 16 + l16];
    B1.y = xB[(long)(na + 1) * Cc + c0 + 16 + l16];
    a00 = wmma_f32(A0, B0, a00);
    a01 = wmma_f32(A0, B1, a01);
    a10 = wmma_f32(A1, B0, a10);
    a11 = wmma_f32(A1, B1, a11);
  }

#pragma unroll
  for (int r = 0; r < 8; ++r) {
    const int m = r + 8 * lhi;
    atomicAdd(&kvLds[m * Dd + l16],             a00[r]);
    atomicAdd(&kvLds[m * Dd + 16 + l16],        a01[r]);
    atomicAdd(&kvLds[(16 + m) * Dd + l16],      a10[r]);
    atomicAdd(&kvLds[(16 + m) * Dd + 16 + l16], a11[r]);
  }
  __syncthreads();

  float* kvB = kv + (long)bh * (Dd * Dd);
  for (int i = tid; i < Dd * Dd; i += 256) atomicAdd(&kvB[i], kvLds[i]);
  if (tid < Dd) atomicAdd(&ksum[b * Cc + c0 + tid], ksumLds[tid]);
}

// ---------------------------------------------------------------------------
// K3: out[b,n,h*32+e] = (rope(Qe) @ kv) * z + LePE(conv3x3_dw(x)) + lepe_b
// grid = 96 bh * 32 row-tiles of 128; wave tile 16 rows x 32 cols; K=32.
// ---------------------------------------------------------------------------
__global__ void __launch_bounds__(256) out_kernel(
    const float* __restrict__ x, const float* __restrict__ Qe,
    const float* __restrict__ cosT, const float* __restrict__ sinT,
    const float* __restrict__ ksum, const float* __restrict__ kv,
    const float* __restrict__ lepe_w, const float* __restrict__ lepe_b,
    float* __restrict__ out) {
  __shared__ float zbuf[128];
  const int tid = threadIdx.x, wave = tid >> 5, lane = tid & 31;
  const int l16 = lane & 15, lhi = lane >> 4;
  const int bid = blockIdx.x;
  const int ntile = bid & 31;
  const int bh = bid >> 5;
  const int b = bh / NHh, h = bh % NHh, c0 = h * Dd;
  const float* QeB = Qe + (long)b * Nn * Cc;
  const float* xB = x + (long)b * Nn * Cc;
  const float* kvB = kv + (long)bh * (Dd * Dd);
  const int n0 = ntile * 128 + wave * 16;

  // z per row (16 lanes of each wave)
  if (lane < 16) {
    const int n = n0 + lane;
    const float* qp = QeB + (long)n * Cc + c0;
    const float* ks = ksum + b * Cc + c0;
    float dot = 0.0f;
#pragma unroll
    for (int i = 0; i < Dd; ++i) dot += qp[i] * ks[i];
    zbuf[wave * 16 + lane] = 1.0f / (dot * (1.0f / (float)Nn) + 1e-6f);
  }

  // WMMA: A = rope(Qe) (16 x 32), B = kv (32 x 32)
  v8f acc0 = {}, acc1 = {};
  const int n = n0 + l16;
#pragma unroll
  for (int k0 = 0; k0 < Dd; k0 += 4) {
    const int d = k0 + 2 * lhi;  // even -> aligned rope pair (d, d+1)
    v2f eo = *(const v2f*)(QeB + (long)n * Cc + c0 + d);
    const int j = (c0 + d) >> 1;
    const float cs = cosT[n * 96 + j];
    const float sn = sinT[n * 96 + j];
    v2f A, B0, B1;
    A.x = eo.x * cs - eo.y * sn;  // rotated[d]
    A.y = eo.x * sn + eo.y * cs;  // rotated[d+1]
    B0.x = kvB[d * Dd + l16];
    B0.y = kvB[(d + 1) * Dd + l16];
    B1.x = kvB[d * Dd + 16 + l16];
    B1.y = kvB[(d + 1) * Dd + 16 + l16];
    acc0 = wmma_f32(A, B0, acc0);
    acc1 = wmma_f32(A, B1, acc1);
  }
  __syncthreads();

  // epilogue: z scaling + depthwise 3x3 conv + bias
#pragma unroll
  for (int t = 0; t < 2; ++t) {
    v8f acc = t ? acc1 : acc0;
    const int c = c0 + t * 16 + l16;
    float w9[9];
#pragma unroll
    for (int i = 0; i < 9; ++i) w9[i] = lepe_w[c * 9 + i];
    const float cbias = lepe_b[c];
#pragma unroll
    for (int r = 0; r < 8; ++r) {
      const int m = r + 8 * lhi;
      const int nr = n0 + m;
      const float z = zbuf[wave * 16 + m];
      const int hp = nr >> 6, wp = nr & 63;
      float conv = cbias;
#pragma unroll
      for (int di = 0; di < 3; ++di) {
        const int hh = hp + di - 1;
        if (hh < 0 || hh >= Hh) continue;
#pragma unroll
        for (int dj = 0; dj < 3; ++dj) {
          const int ww = wp + dj - 1;
          if (ww < 0 || ww >= Ww) continue;
          conv += xB[((long)hh * Ww + ww) * Cc + c] * w9[di * 3 + dj];
        }
      }
      out[((long)b * Nn + nr) * Cc + c] = acc[r] * z + conv;
    }
  }
}

// ---------------------------------------------------------------------------
extern "C" void kernel_launch(void* const* d_in, const int* in_sizes, int n_in,
                              void* d_out, int out_size, void* d_ws,
                              size_t ws_size, hipStream_t stream) {
  const float* x      = (const float*)d_in[0];
  const float* qk_w   = (const float*)d_in[1];
  const float* qk_b   = (const float*)d_in[2];
  const float* lepe_w = (const float*)d_in[3];
  const float* lepe_b = (const float*)d_in[4];
  float* out = (float*)d_out;

  float* ws   = (float*)d_ws;
  float* cosT = ws;                        // 393216
  float* sinT = cosT + Hh * Ww * 96;       // 393216
  float* Qe   = sinT + Hh * Ww * 96;       // 12582912
  float* Ke   = Qe + (long)Bb * Nn * Cc;   // 12582912
  float* ksum = Ke + (long)Bb * Nn * Cc;   // 3072
  float* kvb  = ksum + Bb * Cc;            // 98304

  init_kernel<<<(Hh * Ww * 96 + 255) / 256, 256, 0, stream>>>(cosT, sinT,
                                                              ksum, kvb);
  dim3 g1(Bb * Nn / 64, (2 * Cc) / 64);    // 1024 x 6
  qk_gemm_kernel<<<g1, 256, 0, stream>>>(x, qk_w, qk_b, Qe, Ke);
  dim3 g2(Bb * NHh, 8);                    // 96 x 8
  kv_kernel<<<g2, 256, 0, stream>>>(x, Ke, cosT, sinT, ksum, kvb);
  out_kernel<<<Bb * NHh * 32, 256, 0, stream>>>(x, Qe, cosT, sinT, ksum, kvb,
                                                lepe_w, lepe_b, out);
}